// NTMMemory_56521769615913
// MI455X (gfx1250) — compile-verified
//
#include <hip/hip_runtime.h>

// NTM memory fused kernel for gfx1250 (MI455X).
// One workgroup per batch. Weight pipeline in LDS; reading GEMM via
// V_WMMA_F32_16X16X4_F32 (f32 in/out, matches reference precision class).

typedef __attribute__((ext_vector_type(2))) float v2f;
typedef __attribute__((ext_vector_type(8))) float v8f;

#define Bb 128
#define Hh 4
#define Nn 8192
#define Mm 64
#define Ssz 3
#define EPSF 1e-8f
#define NTHREADS 512
#define NWAVES 16

__device__ __forceinline__ float wave_sum32(float v) {
  #pragma unroll
  for (int o = 16; o > 0; o >>= 1) v += __shfl_xor(v, o, 32);
  return v;
}
__device__ __forceinline__ float wave_max32(float v) {
  #pragma unroll
  for (int o = 16; o > 0; o >>= 1) v = fmaxf(v, __shfl_xor(v, o, 32));
  return v;
}

extern "C" __global__ void __launch_bounds__(NTHREADS)
ntm_fused_kernel(const float* __restrict__ memory, const float* __restrict__ kk,
                 const float* __restrict__ beta, const float* __restrict__ pw,
                 const float* __restrict__ gg_in, const float* __restrict__ sh,
                 const float* __restrict__ gamma, const float* __restrict__ erase,
                 const float* __restrict__ addv, float* __restrict__ out) {
  extern __shared__ float smem[];
  float* w1 = smem;             // [4][8192]: logits -> exp -> wg ; later WMMA partial tiles
  float* w2 = smem + Hh * Nn;   // [4][8192]: wgam -> wl
  float* aux   = smem + 2 * Hh * Nn;
  float* kv    = aux;           // 256: keys
  float* kn    = aux + 256;     // 4: key norms
  float* bet   = aux + 260;     // 4
  float* gg    = aux + 264;     // 4
  float* ginv  = aux + 268;     // 4
  float* gam   = aux + 272;     // 4
  float* sv    = aux + 276;     // 12: shift taps
  float* erz   = aux + 288;     // 64: erase (last head)
  float* adz   = aux + 352;     // 64: add (last head)
  float* hmax  = aux + 416;     // 4
  float* hsum  = aux + 420;     // 4
  float* hsum2 = aux + 424;     // 4
  float* s3v   = aux + 428;     // 4
  float* wred  = aux + 432;     // 64: per-wave reduction slots
  float* Pm    = aux + 496;     // 512: P1(rows0-3)/P2(rows4-7) 8x64

  const int b    = blockIdx.x;
  const int tid  = threadIdx.x;
  const int lane = tid & 31;
  const int wave = tid >> 5;

  // ---------------- setup: stage small tensors into LDS ----------------
  if (tid < Hh * Mm) kv[tid] = kk[(size_t)b * Hh * Mm + tid];
  if (tid < Hh) {
    bet[tid] = beta[b * Hh + tid];
    float gval = gg_in[b * Hh + tid];
    gg[tid] = gval;
    ginv[tid] = 1.0f - gval;
    gam[tid] = gamma[b * Hh + tid];
    for (int j = 0; j < Ssz; ++j) sv[tid * 3 + j] = sh[(b * Hh + tid) * Ssz + j];
  }
  if (tid < Mm) {
    erz[tid] = erase[((size_t)b * Hh + (Hh - 1)) * Mm + tid];
    adz[tid] = addv[((size_t)b * Hh + (Hh - 1)) * Mm + tid];
  }
  __syncthreads();
  if (tid < Hh) {
    float ss = 0.0f;
    for (int m = 0; m < Mm; ++m) { float x = kv[tid * Mm + m]; ss += x * x; }
    kn[tid] = sqrtf(ss);
  }
  __syncthreads();

  // ---------------- phase 1: logits t = beta * cossim -------------------
  // one wave per memory row; lane holds cols {2l, 2l+1} (coalesced float2)
  {
    const float2* mem2 = reinterpret_cast<const float2*>(memory + (size_t)b * Nn * Mm);
    const float kx0 = kv[0 * Mm + 2 * lane], ky0 = kv[0 * Mm + 2 * lane + 1];
    const float kx1 = kv[1 * Mm + 2 * lane], ky1 = kv[1 * Mm + 2 * lane + 1];
    const float kx2 = kv[2 * Mm + 2 * lane], ky2 = kv[2 * Mm + 2 * lane + 1];
    const float kx3 = kv[3 * Mm + 2 * lane], ky3 = kv[3 * Mm + 2 * lane + 1];
    for (int n = wave; n < Nn; n += NWAVES) {
      float2 v = mem2[n * 32 + lane];
      float ssq = v.x * v.x + v.y * v.y;
      float d0 = kx0 * v.x + ky0 * v.y;
      float d1 = kx1 * v.x + ky1 * v.y;
      float d2 = kx2 * v.x + ky2 * v.y;
      float d3 = kx3 * v.x + ky3 * v.y;
      ssq = wave_sum32(ssq);
      d0 = wave_sum32(d0); d1 = wave_sum32(d1);
      d2 = wave_sum32(d2); d3 = wave_sum32(d3);
      if (lane == 0) {
        float mn = sqrtf(ssq);
        w1[0 * Nn + n] = bet[0] * d0 / fmaxf(kn[0] * mn, EPSF);
        w1[1 * Nn + n] = bet[1] * d1 / fmaxf(kn[1] * mn, EPSF);
        w1[2 * Nn + n] = bet[2] * d2 / fmaxf(kn[2] * mn, EPSF);
        w1[3 * Nn + n] = bet[3] * d3 / fmaxf(kn[3] * mn, EPSF);
      }
    }
  }
  __syncthreads();

  // ---------------- phase 2: softmax max over N --------------------------
  for (int h = 0; h < Hh; ++h) {
    float mx = -INFINITY;
    for (int n = tid; n < Nn; n += NTHREADS) mx = fmaxf(mx, w1[h * Nn + n]);
    mx = wave_max32(mx);
    if (lane == 0) wred[wave * 4 + h] = mx;
  }
  __syncthreads();
  if (tid < Hh) {
    float mx = -INFINITY;
    for (int wv = 0; wv < NWAVES; ++wv) mx = fmaxf(mx, wred[wv * 4 + tid]);
    hmax[tid] = mx;
  }
  __syncthreads();

  // ---------------- phase 3: exp + sum -----------------------------------
  for (int h = 0; h < Hh; ++h) {
    const float mxh = hmax[h];
    float acc = 0.0f;
    for (int n = tid; n < Nn; n += NTHREADS) {
      float e = __expf(w1[h * Nn + n] - mxh);
      w1[h * Nn + n] = e;
      acc += e;
    }
    acc = wave_sum32(acc);
    if (lane == 0) wred[wave * 4 + h] = acc;
  }
  __syncthreads();
  if (tid < Hh) {
    float s = 0.0f;
    for (int wv = 0; wv < NWAVES; ++wv) s += wred[wv * 4 + tid];
    hsum[tid] = s;
  }
  __syncthreads();

  // ---------------- phase 4: wc -> wg (interpolate with pw) --------------
  for (int h = 0; h < Hh; ++h) {
    const float inv = 1.0f / hsum[h];
    const float gv = gg[h], giv = ginv[h];
    const float* pwh = pw + ((size_t)b * Hh + h) * Nn;
    for (int n = tid; n < Nn; n += NTHREADS) {
      float wc = w1[h * Nn + n] * inv;
      w1[h * Nn + n] = gv * wc + giv * pwh[n];
    }
  }
  __syncthreads();

  // ---------------- phase 5: circular shift conv + sharpen ---------------
  // ws[n] = s0*wg[n-1] + s1*wg[n] + s2*wg[n+1]; wgam = ws^gamma
  for (int h = 0; h < Hh; ++h) {
    const float sA = sv[h * 3 + 0], sB = sv[h * 3 + 1], sC = sv[h * 3 + 2];
    const float gmv = gam[h];
    float acc = 0.0f;
    for (int n = tid; n < Nn; n += NTHREADS) {
      int nm = (n == 0) ? (Nn - 1) : (n - 1);
      int np = (n == Nn - 1) ? 0 : (n + 1);
      float ws = sA * w1[h * Nn + nm] + sB * w1[h * Nn + n] + sC * w1[h * Nn + np];
      float wgam = (ws > 0.0f) ? __expf(gmv * __logf(ws)) : 0.0f;
      w2[h * Nn + n] = wgam;
      acc += wgam;
    }
    acc = wave_sum32(acc);
    if (lane == 0) wred[wave * 4 + h] = acc;
  }
  __syncthreads();
  if (tid < Hh) {
    float s = 0.0f;
    for (int wv = 0; wv < NWAVES; ++wv) s += wred[wv * 4 + tid];
    hsum2[tid] = s;
  }
  __syncthreads();

  // ---------------- phase 6: normalize -> wl; S3[h] = sum wl[h]*wl[3] ----
  for (int h = 0; h < Hh; ++h) {
    const float inv = 1.0f / hsum2[h];
    for (int n = tid; n < Nn; n += NTHREADS) w2[h * Nn + n] *= inv;
  }
  __syncthreads();
  {
    float p0 = 0, p1 = 0, p2 = 0, p3 = 0;
    for (int n = tid; n < Nn; n += NTHREADS) {
      float w3 = w2[3 * Nn + n];
      p0 += w2[0 * Nn + n] * w3;
      p1 += w2[1 * Nn + n] * w3;
      p2 += w2[2 * Nn + n] * w3;
      p3 += w3 * w3;
    }
    p0 = wave_sum32(p0); p1 = wave_sum32(p1);
    p2 = wave_sum32(p2); p3 = wave_sum32(p3);
    if (lane == 0) {
      wred[wave * 4 + 0] = p0; wred[wave * 4 + 1] = p1;
      wred[wave * 4 + 2] = p2; wred[wave * 4 + 3] = p3;
    }
  }
  __syncthreads();
  if (tid < Hh) {
    float s = 0.0f;
    for (int wv = 0; wv < NWAVES; ++wv) s += wred[wv * 4 + tid];
    s3v[tid] = s;
  }
  __syncthreads();

  // ---------------- phase 7: reading GEMM via V_WMMA_F32_16X16X4_F32 -----
  // A (16xK): rows 0-3 = wl[h], rows 4-7 = wl[h]*wlast, rows 8-15 = 0
  // B (Kx64): raw memory tile (nmemory folded out algebraically)
  {
    const float* memb = memory + (size_t)b * Nn * Mm;
    v8f c0 = {}, c1 = {}, c2 = {}, c3 = {};
    const int col = lane & 15;                 // A row index == B col index
    const int kkof = (lane < 16) ? 0 : 2;      // K slots {0,1} vs {2,3}
    const int r4 = col & 3;
    for (int chunk = wave; chunk < Nn / 4; chunk += NWAVES) {
      const int n0 = chunk * 4;
      const int na = n0 + kkof;
      // branchless A feed (keeps EXEC uniform around WMMA)
      float base0 = w2[r4 * Nn + na];
      float base1 = w2[r4 * Nn + na + 1];
      float wl3a  = w2[3 * Nn + na];
      float wl3b  = w2[3 * Nn + na + 1];
      float m0f = (col < 4) ? 1.0f : wl3a;
      float m1f = (col < 4) ? 1.0f : wl3b;
      v2f av;
      av.x = (col < 8) ? base0 * m0f : 0.0f;
      av.y = (col < 8) ? base1 * m1f : 0.0f;
      const float* mrow = memb + (size_t)na * Mm + col;
      v2f bv;
      bv.x = mrow[0];       bv.y = mrow[Mm];
      c0 = __builtin_amdgcn_wmma_f32_16x16x4_f32(false, av, false, bv, (short)0, c0, false, false);
      bv.x = mrow[16];      bv.y = mrow[16 + Mm];
      c1 = __builtin_amdgcn_wmma_f32_16x16x4_f32(false, av, false, bv, (short)0, c1, false, false);
      bv.x = mrow[32];      bv.y = mrow[32 + Mm];
      c2 = __builtin_amdgcn_wmma_f32_16x16x4_f32(false, av, false, bv, (short)0, c2, false, false);
      bv.x = mrow[48];      bv.y = mrow[48 + Mm];
      c3 = __builtin_amdgcn_wmma_f32_16x16x4_f32(false, av, false, bv, (short)0, c3, false, false);
    }
    // spill per-wave C tiles to LDS (w1 is free now): [wave][tile][16x16]
    float* dst = w1 + wave * 1024;
    const int rbase = (lane < 16) ? 0 : 8;
    #pragma unroll
    for (int j = 0; j < 8; ++j) {
      const int ro = (rbase + j) * 16 + col;
      dst[0 * 256 + ro] = c0[j];
      dst[1 * 256 + ro] = c1[j];
      dst[2 * 256 + ro] = c2[j];
      dst[3 * 256 + ro] = c3[j];
    }
  }
  __syncthreads();

  // ---------------- phase 8: cross-wave reduce + combine -----------------
  for (int p = tid; p < 1024; p += NTHREADS) {
    const int tile = p >> 8;
    const int idx = p & 255;
    const int r = idx >> 4;
    const int c = idx & 15;
    if (r < 8) {
      float s = 0.0f;
      for (int wv = 0; wv < NWAVES; ++wv) s += w1[wv * 1024 + p];
      Pm[r * 64 + tile * 16 + c] = s;
    }
  }
  __syncthreads();
  if (tid < Hh * Mm) {
    const int h = tid >> 6, m = tid & 63;
    out[((size_t)b * Hh + h) * Mm + m] =
        Pm[h * 64 + m] - erz[m] * Pm[(4 + h) * 64 + m] + adz[m] * s3v[h];
  }
}

extern "C" void kernel_launch(void* const* d_in, const int* in_sizes, int n_in,
                              void* d_out, int out_size, void* d_ws, size_t ws_size,
                              hipStream_t stream) {
  (void)in_sizes; (void)n_in; (void)out_size; (void)d_ws; (void)ws_size;
  const float* memory = (const float*)d_in[0];
  const float* k      = (const float*)d_in[1];
  const float* beta   = (const float*)d_in[2];
  const float* pw     = (const float*)d_in[3];
  const float* g      = (const float*)d_in[4];
  const float* s      = (const float*)d_in[5];
  const float* gamma  = (const float*)d_in[6];
  const float* erase  = (const float*)d_in[7];
  const float* add    = (const float*)d_in[8];
  float* out = (float*)d_out;

  const size_t smemBytes = (size_t)(2 * Hh * Nn + 1024) * sizeof(float); // ~260 KB of 320 KB WGP LDS
  ntm_fused_kernel<<<dim3(Bb), dim3(NTHREADS), smemBytes, stream>>>(
      memory, k, beta, pw, g, s, gamma, erase, add, out);
}